// YOLOLayer_81784767251080
// MI455X (gfx1250) — compile-verified
//
#include <hip/hip_runtime.h>

// ---------------- problem constants (from reference) ----------------
#define GSZ   128
#define BSZ   64
#define ASZ   3
#define NBOX  1280
#define INV_STRIDE (1.0f / 8.0f)          // stride = 1024/128 = 8
#define TOTAL_ELEMS (BSZ * GSZ * GSZ * ASZ * 5)   // 15,728,640 floats
#define NCHUNK (TOTAL_ELEMS / 4)                   // 3,932,160 float4 chunks
#define RED_BLOCKS 1024
#define RED_TPB    256

typedef float v2f __attribute__((ext_vector_type(2)));
typedef float v8f __attribute__((ext_vector_type(8)));
typedef int   v4i __attribute__((vector_size(16)));
typedef __attribute__((address_space(1))) v4i* as1_v4i_ptr;
typedef __attribute__((address_space(3))) v4i* as3_v4i_ptr;

// ---------------- CDNA5 feature detection ----------------
#if defined(__has_builtin)
#  if __has_builtin(__builtin_amdgcn_global_load_async_to_lds_b128)
#    define HAVE_ASYNC_LDS 1
#  endif
#  if __has_builtin(__builtin_amdgcn_s_wait_asynccnt)
#    define HAVE_WAIT_ASYNC 1
#  endif
#  if __has_builtin(__builtin_amdgcn_wmma_f32_16x16x4_f32)
#    define HAVE_WMMA_F32X4 1
#  endif
#endif
#ifndef HAVE_ASYNC_LDS
#  define HAVE_ASYNC_LDS 0
#endif
#ifndef HAVE_WAIT_ASYNC
#  define HAVE_WAIT_ASYNC 0
#endif
#ifndef HAVE_WMMA_F32X4
#  define HAVE_WMMA_F32X4 0
#endif

#if HAVE_WAIT_ASYNC
#  define WAIT_ASYNC(n) __builtin_amdgcn_s_wait_asynccnt(n)
#else
#  define WAIT_ASYNC(n) asm volatile("s_wait_asynccnt %0" ::"i"(n) : "memory")
#endif

#if HAVE_ASYNC_LDS
__device__ __forceinline__ void async_copy_b128(const void* gsrc, void* ldst) {
  __builtin_amdgcn_global_load_async_to_lds_b128(
      (as1_v4i_ptr)(v4i*)const_cast<void*>(gsrc),
      (as3_v4i_ptr)(v4i*)ldst,
      /*imm offset*/ 0, /*cpol*/ 0);
}
#endif

// ---------------- math helpers ----------------
__device__ __forceinline__ float softplus_f(float x) {
  // log(1 + exp(x)), numerically stable
  return fmaxf(x, 0.0f) + log1pf(__expf(-fabsf(x)));
}
__device__ __forceinline__ float sigmoid_f(float x) {
  return 1.0f / (1.0f + __expf(-x));
}

// per-box: grid cell + best anchor (argmax IoU, first-wins on ties like jnp.argmax)
__device__ __forceinline__ void box_cell(const float* __restrict__ gbb, int n,
                                         const float* __restrict__ anc,
                                         int& gi, int& gj, int& best,
                                         float& gx, float& gy, float& gw, float& gh) {
  gx = gbb[n * 4 + 0] * INV_STRIDE;
  gy = gbb[n * 4 + 1] * INV_STRIDE;
  gw = gbb[n * 4 + 2] * INV_STRIDE;
  gh = gbb[n * 4 + 3] * INV_STRIDE;
  gi = (int)gx;
  gj = (int)gy;
  float bestIou = -1.0f;
  best = 0;
  for (int a = 0; a < ASZ; ++a) {
    float aw = anc[a * 2 + 0] * INV_STRIDE;
    float ah = anc[a * 2 + 1] * INV_STRIDE;
    float inter = fminf(aw, gw) * fminf(ah, gh);
    float uni = aw * ah + gw * gh - inter;
    float iou = inter / uni;
    if (iou > bestIou) { bestIou = iou; best = a; }
  }
}

// ---------------- kernel 1: sparse per-target-cell corrections ----------------
// contribution = 5*coord^2 + softplus(-conf) - 0.5*softplus(conf)  (last-write-wins dedupe)
__global__ void box_loss_kernel(const float* __restrict__ yp,
                                const float* __restrict__ gbb,
                                const float* __restrict__ anc,
                                const int* __restrict__ bidx,
                                float* __restrict__ boxout) {
  int n = blockIdx.x * blockDim.x + threadIdx.x;
  if (n >= NBOX) return;
  int gi, gj, best;
  float gx, gy, gw, gh;
  box_cell(gbb, n, anc, gi, gj, best, gx, gy, gw, gh);
  int b = bidx[n];

  // scatter .set semantics: a later box writing the same (b,best,gj,gi) wins
  bool active = true;
  for (int n2 = n + 1; n2 < NBOX; ++n2) {
    int b2 = bidx[n2];
    if (b2 > b) break;          // batch_idx is sorted ascending
    if (b2 != b) continue;
    int gi2, gj2, best2;
    float x2, y2, w2, h2;
    box_cell(gbb, n2, anc, gi2, gj2, best2, x2, y2, w2, h2);
    if (gi2 == gi && gj2 == gj && best2 == best) { active = false; break; }
  }

  float contrib = 0.0f;
  if (active) {
    float aw = anc[best * 2 + 0] * INV_STRIDE;
    float ah = anc[best * 2 + 1] * INV_STRIDE;
    float tx = gx - (float)gi;
    float ty = gy - (float)gj;
    float tw = logf(gw / aw);
    float th = logf(gh / ah);
    size_t base = ((((size_t)b * GSZ + (size_t)gj) * GSZ + (size_t)gi) * (ASZ * 5)) + (size_t)best * 5;
    float cf = yp[base + 0];
    float dx = sigmoid_f(yp[base + 1]) - tx;
    float dy = sigmoid_f(yp[base + 2]) - ty;
    float dw = yp[base + 3] - tw;
    float dh = yp[base + 4] - th;
    contrib = 5.0f * (dx * dx + dy * dy + dw * dw + dh * dh)
            + softplus_f(-cf)            // + OBJ_L * -log_sigmoid(conf)
            - 0.5f * softplus_f(cf);     // remove this cell's noobj term
  }
  boxout[n] = contrib;
}

// ---------------- kernel 2: stream 63MB, 0.5*sum softplus(conf) -------------
// Async global->LDS double-buffered staging; each lane stages & consumes its own 16B.
// Wave reduction via v_wmma_f32_16x16x4_f32 with all-ones B (D = row-sums of A).
__global__ __launch_bounds__(RED_TPB) void conf_reduce_kernel(
    const float* __restrict__ yp, float* __restrict__ partials) {
  const int tid = threadIdx.x;
  const int gstride = RED_BLOCKS * RED_TPB;
  int i = blockIdx.x * RED_TPB + tid;
  float acc = 0.0f;
  const float4* __restrict__ src = (const float4*)yp;

#if HAVE_ASYNC_LDS
  __shared__ float4 stage[2][RED_TPB];
  int buf = 0;
  if (i < NCHUNK) async_copy_b128(src + i, &stage[0][tid]);
  for (; i < NCHUNK; i += gstride) {
    int nx = i + gstride;
    if (nx < NCHUNK) {
      async_copy_b128(src + nx, &stage[buf ^ 1][tid]);
      WAIT_ASYNC(1);            // oldest (current buf) is done, next in flight
    } else {
      WAIT_ASYNC(0);
    }
    float4 v = stage[buf][tid];
#else
  for (; i < NCHUNK; i += gstride) {
    float4 v = src[i];
#endif
    // elements e..e+3, conf logits sit at flat index % 5 == 0
    int e = i * 4;
    int r = e - (e / 5) * 5;
    int j0 = (5 - r) % 5;       // r==0 -> j0=0 ; r==1 -> j0=4 (no conf in this chunk)
    if (j0 < 4) {
      float c = (j0 == 0) ? v.x : (j0 == 1) ? v.y : (j0 == 2) ? v.z : v.w;
      acc += softplus_f(c);
    }
#if HAVE_ASYNC_LDS
    buf ^= 1;
#endif
  }

  // ---- wave32 reduction ----
  float s;
#if HAVE_WMMA_F32X4
  {
    v2f a;  a[0] = acc;  a[1] = 0.0f;      // A holds each lane's partial once
    v2f bm; bm[0] = 1.0f; bm[1] = 1.0f;    // B = all ones -> D[m,n] = sum_k A[m,k]
    v8f c = {};
    v8f d = __builtin_amdgcn_wmma_f32_16x16x4_f32(
        false, a, false, bm, (short)0, c, false, false);
    s = ((d[0] + d[1]) + (d[2] + d[3])) + ((d[4] + d[5]) + (d[6] + d[7]));
  }
#else
  s = acc;
  for (int off = 1; off < 16; off <<= 1) s += __shfl_xor(s, off, 32);
#endif
  s += __shfl_xor(s, 16, 32);              // combine the two half-wave column sums

  __shared__ float wsum[RED_TPB / 32];
  int lane = tid & 31, wv = tid >> 5;
  if (lane == 0) wsum[wv] = s;
  __syncthreads();
  if (tid == 0) {
    float t = 0.0f;
    for (int w = 0; w < RED_TPB / 32; ++w) t += wsum[w];   // fixed order
    partials[blockIdx.x] = 0.5f * t;       // NOOBJ_L applied to the dense term
  }
}

// ---------------- kernel 3: deterministic final reduce ----------------
__global__ void final_reduce_kernel(const float* __restrict__ vals, int n,
                                    float* __restrict__ out) {
  __shared__ float sm[256];
  float a = 0.0f;
  for (int i = threadIdx.x; i < n; i += 256) a += vals[i];
  sm[threadIdx.x] = a;
  __syncthreads();
  for (int off = 128; off > 0; off >>= 1) {
    if (threadIdx.x < off) sm[threadIdx.x] += sm[threadIdx.x + off];
    __syncthreads();
  }
  if (threadIdx.x == 0) out[0] = sm[0];
}

// ---------------- host entry ----------------
extern "C" void kernel_launch(void* const* d_in, const int* in_sizes, int n_in,
                              void* d_out, int out_size, void* d_ws, size_t ws_size,
                              hipStream_t stream) {
  (void)in_sizes; (void)n_in; (void)out_size; (void)ws_size;
  const float* yp   = (const float*)d_in[0];   // y_pred (B,G,G,A*5)
  const float* gbb  = (const float*)d_in[1];   // ground_bboxes (N,4)
  const float* anc  = (const float*)d_in[2];   // anchors (3,2)
  const int*   bidx = (const int*)d_in[3];     // batch_idx (N,)
  float* out = (float*)d_out;
  float* ws  = (float*)d_ws;

  float* partials = ws;                 // [0, RED_BLOCKS)
  float* boxvals  = ws + RED_BLOCKS;    // [RED_BLOCKS, RED_BLOCKS+NBOX)

  box_loss_kernel<<<(NBOX + 255) / 256, 256, 0, stream>>>(yp, gbb, anc, bidx, boxvals);
  conf_reduce_kernel<<<RED_BLOCKS, RED_TPB, 0, stream>>>(yp, partials);
  final_reduce_kernel<<<1, 256, 0, stream>>>(ws, RED_BLOCKS + NBOX, out);
}